// SocialInteraction4_16716012716118
// MI455X (gfx1250) — compile-verified
//
#include <hip/hip_runtime.h>
#include <hip/hip_bf16.h>

// Problem sizes (fixed by the reference)
#define PED   1024   // P
#define R_DIM 128
#define M_DIM 128

typedef __attribute__((ext_vector_type(2))) float v2f;
typedef __attribute__((ext_vector_type(8))) float v8f;

// ---------------------------------------------------------------------------
// Kernel 1: per-row scores + masked softmax -> weights[P,P] (f32, in d_ws)
//
// One block per row i. 256 threads = 8 waves (wave32).
// Each wave computes score[j] = dot(rela[i,j,:], att_w) cooperatively:
//   lane L holds att_w[4L..4L+3] in registers, loads rela[i,j,4L..4L+3]
//   (coalesced 512B per wave), dot4, then a 5-step wave32 xor-shuffle reduce.
// Then the block reproduces the reference masking/softmax exactly:
//   pos[j]     = nei[i,j] > 0 ? score[j]+b : -1e-6
//   soft       = softmax(pos) over ALL j (max-subtracted, like jax.nn.softmax)
//   weight[j]  = nei[i,j] > 0 ? soft[j] : 0
// ---------------------------------------------------------------------------
__global__ void __launch_bounds__(256)
si_score_softmax_kernel(const float* __restrict__ rela,   // [P,P,R]
                        const int*   __restrict__ nei,    // [P,P]
                        const float* __restrict__ att_w,  // [R]
                        const float* __restrict__ att_b,  // [1]
                        float* __restrict__ Wout)         // [P,P]
{
    __shared__ float s_scores[PED];
    __shared__ float s_red[8];

    const int i    = blockIdx.x;
    const int tid  = threadIdx.x;        // 0..255
    const int lane = tid & 31;
    const int wave = tid >> 5;           // 0..7

    // att_w fragment: lane holds elements [4*lane, 4*lane+3]
    const float4 wf   = *(const float4*)(att_w + lane * 4);
    const float  bias = att_b[0];

    const float* rrow = rela + (size_t)i * PED * R_DIM;

    // Wave-cooperative dot products, 128 j's per wave.
    for (int j = wave; j < PED; j += 8) {
        const float4 rv =
            *(const float4*)(rrow + (size_t)j * R_DIM + lane * 4);
        float acc = rv.x * wf.x + rv.y * wf.y + rv.z * wf.z + rv.w * wf.w;
        #pragma unroll
        for (int off = 16; off >= 1; off >>= 1)
            acc += __shfl_xor(acc, off, 32);
        if (lane == 0) s_scores[j] = acc + bias;
    }
    __syncthreads();

    // Mask -> pos, thread-local over 4 strided j's, then block max.
    const int* nrow = nei + (size_t)i * PED;
    float lpos[4];
    int   lnei[4];
    float lmax = -3.402823466e38f;
    #pragma unroll
    for (int t = 0; t < 4; ++t) {
        const int j  = tid + t * 256;
        const int nv = nrow[j];
        const float p = (nv > 0) ? s_scores[j] : -1e-6f;
        lpos[t] = p;
        lnei[t] = nv;
        lmax = fmaxf(lmax, p);
    }
    #pragma unroll
    for (int off = 16; off >= 1; off >>= 1)
        lmax = fmaxf(lmax, __shfl_xor(lmax, off, 32));
    if (lane == 0) s_red[wave] = lmax;
    __syncthreads();
    float bmax = s_red[0];
    #pragma unroll
    for (int w = 1; w < 8; ++w) bmax = fmaxf(bmax, s_red[w]);
    __syncthreads();   // protect s_red before reuse for the sum

    // exp(pos - max) and block sum.
    float lexp[4];
    float lsum = 0.0f;
    #pragma unroll
    for (int t = 0; t < 4; ++t) {
        lexp[t] = expf(lpos[t] - bmax);
        lsum += lexp[t];
    }
    #pragma unroll
    for (int off = 16; off >= 1; off >>= 1)
        lsum += __shfl_xor(lsum, off, 32);
    if (lane == 0) s_red[wave] = lsum;
    __syncthreads();
    float bsum = 0.0f;
    #pragma unroll
    for (int w = 0; w < 8; ++w) bsum += s_red[w];
    const float inv = 1.0f / bsum;

    // weights = mask ? softmax : 0 ; coalesced stores.
    float* orow = Wout + (size_t)i * PED;
    #pragma unroll
    for (int t = 0; t < 4; ++t) {
        const int j = tid + t * 256;
        orow[j] = (lnei[t] > 0) ? lexp[t] * inv : 0.0f;
    }
}

// ---------------------------------------------------------------------------
// Kernel 2: out[P,M] = weights[P,P] @ hidden[P,M] via V_WMMA_F32_16X16X4_F32.
//
// Grid: 64 blocks (i-tiles), 8 waves/block (m-tiles of 16). Each wave owns a
// 16x16 output tile and iterates K=1024 in steps of 4.
//
// ISA lane layouts (CDNA5 ISA):
//   A (16x4 f32, 2 VGPRs): lanes 0-15 row=lane hold K={k0,k0+1};
//                          lanes 16-31 row=lane-16 hold K={k0+2,k0+3}.
//   B (4x16 f32, 2 VGPRs): lanes 0-15 col=lane, rows {k0,k0+1};
//                          lanes 16-31 col=lane-16, rows {k0+2,k0+3}.
//   C/D (16x16 f32, 8 VGPRs): VGPR v -> row v (lanes 0-15, col=lane)
//                             and row v+8 (lanes 16-31, col=lane-16).
// ---------------------------------------------------------------------------
__global__ void __launch_bounds__(256)
si_wmma_gemm_kernel(const float* __restrict__ W,    // [P,P] softmax weights
                    const float* __restrict__ Hd,   // [P,M] hidden_state
                    float* __restrict__ out)        // [P,M]
{
    const int lane = threadIdx.x & 31;
    const int wave = threadIdx.x >> 5;     // 0..7  -> m tile
    const int i0   = blockIdx.x * 16;      // i tile origin
    const int m0   = wave * 16;            // m tile origin
    const int half = lane >> 4;            // 0: K pair {0,1}; 1: K pair {2,3}
    const int l15  = lane & 15;

    // A: row (i0+l15) of W, starting at column 2*half; float2 per k-step.
    const float* Arow  = W + (size_t)(i0 + l15) * PED + 2 * half;
    // B: column (m0+l15) of Hd, starting at row 2*half.
    const float* Bbase = Hd + (size_t)(2 * half) * M_DIM + (m0 + l15);

    v8f c = {};
    #pragma unroll 8
    for (int k0 = 0; k0 < PED; k0 += 4) {
        const float2 a2 = *(const float2*)(Arow + k0);
        v2f av; av[0] = a2.x; av[1] = a2.y;
        v2f bv;
        bv[0] = Bbase[(size_t)k0 * M_DIM];
        bv[1] = Bbase[(size_t)(k0 + 1) * M_DIM];
        c = __builtin_amdgcn_wmma_f32_16x16x4_f32(
                /*neg_a=*/false, av, /*neg_b=*/false, bv,
                /*c_mod=*/(short)0, c, /*reuse_a=*/false, /*reuse_b=*/false);
    }

    // Store D tile: VGPR v -> rows v (lanes 0-15) and v+8 (lanes 16-31).
    #pragma unroll
    for (int v = 0; v < 8; ++v) {
        const int row = i0 + v + 8 * half;
        out[(size_t)row * M_DIM + m0 + l15] = c[v];
    }
}

extern "C" void kernel_launch(void* const* d_in, const int* in_sizes, int n_in,
                              void* d_out, int out_size, void* d_ws, size_t ws_size,
                              hipStream_t stream) {
    const float* hidden = (const float*)d_in[0];   // [P,M]
    const float* rela   = (const float*)d_in[1];   // [P,P,R]
    // d_in[2] = corr_index [P,P,2] (unused by the reference computation)
    const int*   nei    = (const int*)d_in[3];     // [P,P]
    const float* att_w  = (const float*)d_in[4];   // [R]
    const float* att_b  = (const float*)d_in[5];   // [1]
    float*       outp   = (float*)d_out;           // [P,M]

    float* weights = (float*)d_ws;                 // needs P*P*4 = 4 MB

    si_score_softmax_kernel<<<PED, 256, 0, stream>>>(rela, nei, att_w, att_b,
                                                     weights);
    si_wmma_gemm_kernel<<<PED / 16, 256, 0, stream>>>(weights, hidden, outp);
}